// AttentionLayer_49185965474433
// MI455X (gfx1250) — compile-verified
//
#include <hip/hip_runtime.h>

// ---------------------------------------------------------------------------
// Bahdanau attention for MI455X (gfx1250, wave32).
//   B=16, Te=256, Td=128, H=512  (hardcoded from reference setup_inputs)
// Pipeline:
//   1) Ws = enc @ W_a            (WMMA f32 16x16x4, batched)
//   2) Uh = dec @ U_a            (WMMA f32 16x16x4, batched)
//   3) e  = softmax_t( tanh(Ws+Uh) @ V )   (v_tanh_f32 + wave32 reductions,
//                                           d-tiled 8x to reuse Ws in regs)
//   4) c  = e @ enc              (WMMA f32 16x16x4, batched)
// fp32 end-to-end: GEMM FLOPs (~3.8G) are negligible vs 23.3 TB/s HBM and the
// 268M-element tanh stage, so the fp32 matrix pipe is the right choice.
// ---------------------------------------------------------------------------

typedef __attribute__((ext_vector_type(2))) float v2f;
typedef __attribute__((ext_vector_type(8))) float v8f;

#define ATT_B  16
#define ATT_TE 256
#define ATT_TD 128
#define ATT_H  512
#define ATT_DT 8      // decoder steps per workgroup in the energies kernel

static __device__ __forceinline__ float fast_tanh(float x) {
#if __has_builtin(__builtin_amdgcn_tanhf)
  return __builtin_amdgcn_tanhf(x);   // v_tanh_f32 (CDNA5 TRANS op)
#else
  return tanhf(x);
#endif
}

// ---------------------------------------------------------------------------
// Batched row-major GEMM: C[b] = A[b] * B[b]   (M,N multiples of 16, K of 4)
// One wave computes one 16x16 tile; 4 waves per block span 64 columns.
// ISA 7.12.2 layouts:
//   A (16x4 f32): lanes 0-15 -> rows M=0..15 holding K=0,1; lanes 16-31 K=2,3
//   B (4x16 f32): lanes 0-15 -> cols N=0..15 holding K=0,1; lanes 16-31 K=2,3
//   C/D (16x16 f32): VGPR r -> M = r + 8*(lane/16), N = lane%16
// ---------------------------------------------------------------------------
__global__ void __launch_bounds__(128)
gemm_wmma_f32_kernel(const float* __restrict__ A, const float* __restrict__ B,
                     float* __restrict__ C, int M, int N, int K,
                     long long sA, long long sB, long long sC)
{
  const int b = blockIdx.z;
  A += (long long)b * sA;
  B += (long long)b * sB;
  C += (long long)b * sC;

  const int lane  = threadIdx.x & 31;
  const int wave  = threadIdx.x >> 5;
  const int m0    = blockIdx.y * 16;
  const int n0    = (blockIdx.x * 4 + wave) * 16;
  if (m0 >= M || n0 >= N) return;   // wave-uniform: EXEC stays all-ones for WMMA

  const int row   = lane & 15;      // A row (or B col) within the tile
  const int khalf = lane >> 4;      // selects K pair {0,1} or {2,3}

  const float* aptr = A + (long long)(m0 + row) * K + 2 * khalf;  // 8B aligned
  const float* bcol = B + (n0 + row);

  v8f acc = {};
  for (int k0 = 0; k0 < K; k0 += 4) {
    v2f a = *(const v2f*)(aptr + k0);                  // A[m0+row][k0+2kh .. +1]
    const float* bp = bcol + (long long)(k0 + 2 * khalf) * N;
    v2f bm;
    bm.x = bp[0];                                      // B[k0+2kh  ][n0+row]
    bm.y = bp[N];                                      // B[k0+2kh+1][n0+row]
    acc = __builtin_amdgcn_wmma_f32_16x16x4_f32(
        /*neg_a=*/false, a, /*neg_b=*/false, bm,
        /*c_mod=*/(short)0, acc, /*reuse_a=*/false, /*reuse_b=*/false);
  }

  const int colOut  = n0 + (lane & 15);
  const int rowBase = m0 + (lane >> 4) * 8;
#pragma unroll
  for (int r = 0; r < 8; ++r)
    C[(long long)(rowBase + r) * N + colOut] = acc[r];
}

// ---------------------------------------------------------------------------
// Energies + softmax, d-tiled: one 256-thread block per (b, 8 decoder steps).
//   e[b,d,t] = softmax_t( sum_h tanh(Ws[b,t,h] + Uh[b,d,h]) * V[h] )
// Each wave owns t = wave, wave+8, ...; it loads the Ws row ONCE into 16
// registers (coalesced 128B loads) and reuses it for all 8 d's, whose Uh rows
// live in LDS. V is hoisted into registers (lane->h mapping is t-invariant).
// Cuts Ws L2 traffic 8x vs the naive per-(b,d) scheme; the loop is then
// limited by the v_tanh_f32 TRANS pipe at ~1 op/lane/cycle with one matching
// ds_load_b32 per tanh.
// ---------------------------------------------------------------------------
__global__ void __launch_bounds__(256)
energies_softmax_kernel(const float* __restrict__ Ws,   // [B,Te,H]
                        const float* __restrict__ Uh,   // [B,Td,H]
                        const float* __restrict__ Va,   // [H]
                        float* __restrict__ Eout)       // [B,Td,Te]
{
  __shared__ float s_uh[ATT_DT][ATT_H];   // 16 KB
  __shared__ float s_e[ATT_DT][ATT_TE];   //  8 KB
  __shared__ float s_red[8];

  const int b    = blockIdx.y;
  const int d0   = blockIdx.x * ATT_DT;
  const int tid  = threadIdx.x;
  const int lane = tid & 31;
  const int wave = tid >> 5;

  // Stage the 8 Uh rows for this d-tile.
  for (int i = tid; i < ATT_DT * ATT_H; i += 256) {
    const int dd = i / ATT_H, h = i % ATT_H;
    s_uh[dd][h] = Uh[((long long)b * ATT_TD + d0 + dd) * ATT_H + h];
  }
  // V into registers: h = lane + 32j is the same for every t this lane touches.
  float vreg[ATT_H / 32];
#pragma unroll
  for (int j = 0; j < ATT_H / 32; ++j)
    vreg[j] = Va[lane + 32 * j];
  __syncthreads();

  const float* wsb = Ws + (long long)b * ATT_TE * ATT_H;
  for (int t = wave; t < ATT_TE; t += 8) {
    const float* wr = wsb + (long long)t * ATT_H;
    float wreg[ATT_H / 32];
#pragma unroll
    for (int j = 0; j < ATT_H / 32; ++j)
      wreg[j] = wr[lane + 32 * j];                 // coalesced, loaded once

#pragma unroll
    for (int dd = 0; dd < ATT_DT; ++dd) {
      float acc = 0.f;
#pragma unroll
      for (int j = 0; j < ATT_H / 32; ++j)
        acc += fast_tanh(wreg[j] + s_uh[dd][lane + 32 * j]) * vreg[j];
#pragma unroll
      for (int off = 16; off > 0; off >>= 1)
        acc += __shfl_xor(acc, off, 32);
      if (lane == 0) s_e[dd][t] = acc;
    }
  }
  __syncthreads();

  // Block softmax over each of the 8 rows (thread tid owns column t = tid).
  for (int dd = 0; dd < ATT_DT; ++dd) {
    const float val = s_e[dd][tid];
    float m = val;
#pragma unroll
    for (int off = 16; off > 0; off >>= 1)
      m = fmaxf(m, __shfl_xor(m, off, 32));
    if (lane == 0) s_red[wave] = m;
    __syncthreads();
#pragma unroll
    for (int w = 0; w < 8; ++w) m = fmaxf(m, s_red[w]);

    const float ex = __expf(val - m);
    float s = ex;
#pragma unroll
    for (int off = 16; off > 0; off >>= 1)
      s += __shfl_xor(s, off, 32);
    __syncthreads();                  // done reading s_red (max phase)
    if (lane == 0) s_red[wave] = s;
    __syncthreads();
    float tot = 0.f;
#pragma unroll
    for (int w = 0; w < 8; ++w) tot += s_red[w];

    Eout[((long long)b * ATT_TD + d0 + dd) * ATT_TE + tid] = ex / tot;
    __syncthreads();                  // s_red reused next dd iteration
  }
}

// ---------------------------------------------------------------------------
extern "C" void kernel_launch(void* const* d_in, const int* in_sizes, int n_in,
                              void* d_out, int out_size, void* d_ws, size_t ws_size,
                              hipStream_t stream) {
  (void)in_sizes; (void)n_in; (void)out_size; (void)ws_size;

  const float* enc = (const float*)d_in[0];  // [16,256,512]
  const float* dec = (const float*)d_in[1];  // [16,128,512]
  const float* Wa  = (const float*)d_in[2];  // [512,512]
  const float* Ua  = (const float*)d_in[3];  // [512,512]
  const float* Va  = (const float*)d_in[4];  // [512,1] -> flat [512]

  float* ws_buf = (float*)d_ws;
  float* Ws = ws_buf;                                       // 16*256*512
  float* Uh = ws_buf + (size_t)ATT_B * ATT_TE * ATT_H;      // 16*128*512

  float* c_out = (float*)d_out;                             // [16,128,512]
  float* e_out = c_out + (size_t)ATT_B * ATT_TD * ATT_H;    // [16,128,256]

  // 1) Ws = enc @ W_a   (M=256, N=512, K=512; B shared across batch)
  gemm_wmma_f32_kernel<<<dim3(ATT_H / 64, ATT_TE / 16, ATT_B), 128, 0, stream>>>(
      enc, Wa, Ws, ATT_TE, ATT_H, ATT_H,
      (long long)ATT_TE * ATT_H, 0LL, (long long)ATT_TE * ATT_H);

  // 2) Uh = dec @ U_a   (M=128, N=512, K=512)
  gemm_wmma_f32_kernel<<<dim3(ATT_H / 64, ATT_TD / 16, ATT_B), 128, 0, stream>>>(
      dec, Ua, Uh, ATT_TD, ATT_H, ATT_H,
      (long long)ATT_TD * ATT_H, 0LL, (long long)ATT_TD * ATT_H);

  // 3) e = softmax(tanh(Ws+Uh) @ V), 8 decoder steps per workgroup
  energies_softmax_kernel<<<dim3(ATT_TD / ATT_DT, ATT_B), 256, 0, stream>>>(
      Ws, Uh, Va, e_out);

  // 4) c = e @ enc      (M=128, N=512, K=256; both operands batched)
  gemm_wmma_f32_kernel<<<dim3(ATT_H / 64, ATT_TD / 16, ATT_B), 128, 0, stream>>>(
      e_out, enc, c_out, ATT_TD, ATT_H, ATT_TE,
      (long long)ATT_TD * ATT_TE, (long long)ATT_TE * ATT_H,
      (long long)ATT_TD * ATT_H);
}